// Loss_20933670601009
// MI455X (gfx1250) — compile-verified
//
#include <hip/hip_runtime.h>
#include <math.h>
#include <stdint.h>

// Problem constants from the reference
#define L_A        128
#define BATCH      32
#define NPOS       (L_A * BATCH)     // 4096 positions
#define NUM_RULES  2048
#define NUM_TOKENS 32000
#define REF_LEN    512
#define EPS_F      1e-7f

#define NT    1024                   // one workgroup, 32 wave32 waves
#define ITER  (NPOS / NT)            // 4 positions per thread
#define NSLOT (3 * ITER)             // 12 gathered values per thread

// One CDNA5 async global->LDS gather: per-lane scattered global address,
// per-lane LDS byte offset. Tracked by ASYNCcnt, not LOADcnt.
__device__ __forceinline__ void async_gather_b32(uint32_t lds_byte_off,
                                                 const float* gptr) {
  unsigned long long ga = (unsigned long long)(uintptr_t)gptr;
  asm volatile("global_load_async_to_lds_b32 %0, %1, off"
               :: "v"(lds_byte_off), "v"(ga)
               : "memory");
}

__global__ __launch_bounds__(NT, 1) void loss_gather_log_reduce(
    const float* __restrict__ rule_probs,       // (L_A,B,NUM_RULES)
    const float* __restrict__ token_probs,      // (L_A,B,NUM_TOKENS)
    const float* __restrict__ reference_probs,  // (L_A,B,REF_LEN)
    const int*   __restrict__ gt_actions,       // (L_A,B,3)
    const int*   __restrict__ mask,             // (L_A,B)
    float*       __restrict__ out)              // scalar
{
  // SoA layout: slot j for thread t lives at gbuf[j*NT + t] -> conflict-free ds reads
  __shared__ float gbuf[NSLOT * NT];            // 48 KB
  __shared__ float wave_sums[NT / 32];

  const int t = threadIdx.x;

  int idx_r[ITER], idx_t[ITER], idx_f[ITER], mk[ITER];

  // Phase 0: issue ALL metadata loads back-to-back (independent, coalesced).
  // Keeping this loop free of any use of the loaded values lets the compiler
  // fire all 16 vector loads before the first s_wait_loadcnt.
  #pragma unroll
  for (int i = 0; i < ITER; ++i) {
    const int pos = i * NT + t;
    idx_r[i] = gt_actions[pos * 3 + 0];
    idx_t[i] = gt_actions[pos * 3 + 1];
    idx_f[i] = gt_actions[pos * 3 + 2];
    mk[i]    = mask[pos];
  }

  // Phase 1: fire all 12 scattered gathers per lane, fire-and-forget into LDS.
  #pragma unroll
  for (int i = 0; i < ITER; ++i) {
    const int pos = i * NT + t;
    const int rc  = idx_r[i] < 0 ? 0 : idx_r[i];   // jnp.clip(idx, 0)
    const int tkc = idx_t[i] < 0 ? 0 : idx_t[i];
    const int rfc = idx_f[i] < 0 ? 0 : idx_f[i];

    const uint32_t l0 = (uint32_t)(uintptr_t)&gbuf[(i * 3 + 0) * NT + t];
    const uint32_t l1 = (uint32_t)(uintptr_t)&gbuf[(i * 3 + 1) * NT + t];
    const uint32_t l2 = (uint32_t)(uintptr_t)&gbuf[(i * 3 + 2) * NT + t];

    async_gather_b32(l0, rule_probs      + (size_t)pos * NUM_RULES  + rc);
    async_gather_b32(l1, token_probs     + (size_t)pos * NUM_TOKENS + tkc);
    async_gather_b32(l2, reference_probs + (size_t)pos * REF_LEN    + rfc);
  }

  // Single wait for all 12 outstanding async transfers of this wave.
  asm volatile("s_wait_asynccnt 0x0" ::: "memory");

  // Phase 2: consume own LDS slots (no cross-thread traffic -> no barrier needed),
  // compute -log(prob)*mask, accumulate.
  float local = 0.0f;
  #pragma unroll
  for (int i = 0; i < ITER; ++i) {
    const float g0 = gbuf[(i * 3 + 0) * NT + t];
    const float g1 = gbuf[(i * 3 + 1) * NT + t];
    const float g2 = gbuf[(i * 3 + 2) * NT + t];
    float p = (idx_r[i] >= 0 ? g0 : 0.0f)
            + (idx_t[i] >= 0 ? g1 : 0.0f)
            + (idx_f[i] >= 0 ? g2 : 0.0f);
    p += (p < EPS_F) ? EPS_F : 0.0f;           // exact reference semantics (add, not clamp)
    local -= logf(p) * (float)mk[i];
  }

  // Phase 3: deterministic block reduction (wave32 shuffle -> LDS -> wave32 shuffle).
  #pragma unroll
  for (int o = 16; o > 0; o >>= 1) local += __shfl_down(local, o, 32);
  if ((t & 31) == 0) wave_sums[t >> 5] = local;
  __syncthreads();
  if (t < 32) {
    float v = wave_sums[t];
    #pragma unroll
    for (int o = 16; o > 0; o >>= 1) v += __shfl_down(v, o, 32);
    if (t == 0) out[0] = v * (1.0f / (float)BATCH);  // mean over B of per-B sums
  }
}

extern "C" void kernel_launch(void* const* d_in, const int* in_sizes, int n_in,
                              void* d_out, int out_size, void* d_ws, size_t ws_size,
                              hipStream_t stream) {
  (void)in_sizes; (void)n_in; (void)out_size; (void)d_ws; (void)ws_size;
  const float* rule_probs      = (const float*)d_in[0];
  const float* token_probs     = (const float*)d_in[1];
  const float* reference_probs = (const float*)d_in[2];
  const int*   gt_actions      = (const int*)d_in[3];
  const int*   mask            = (const int*)d_in[4];
  float*       out             = (float*)d_out;

  loss_gather_log_reduce<<<1, NT, 0, stream>>>(
      rule_probs, token_probs, reference_probs, gt_actions, mask, out);
}